// WarpingLayer_SF_PWC_27848567947670
// MI455X (gfx1250) — compile-verified
//
#include <hip/hip_runtime.h>
#include <hip/hip_bf16.h>
#include <stdint.h>

// Problem geometry (from reference setup_inputs)
#define H_IN   64
#define W_IN   208
#define H_OUT  128
#define W_OUT  416
#define C_X    128
#define CGRP   4                 // channel groups per pixel
#define CPG    (C_X / CGRP)      // 32 channels per group
#define HW_OUT (H_OUT * W_OUT)

// ---------------- CDNA5 async global->LDS helpers ----------------
typedef __attribute__((address_space(1))) int* glob_i32_p;
typedef __attribute__((address_space(3))) int* lds_i32_p;

__device__ __forceinline__ void async_copy_b32(const float* g, float* l) {
#if __has_builtin(__builtin_amdgcn_global_load_async_to_lds_b32)
  __builtin_amdgcn_global_load_async_to_lds_b32(
      (glob_i32_p)g, (lds_i32_p)l, /*offset=*/0, /*cpol=*/0);
#else
  uint32_t lds_off = (uint32_t)(uintptr_t)(lds_i32_p)l;
  uint64_t gaddr   = (uint64_t)(uintptr_t)g;
  asm volatile("global_load_async_to_lds_b32 %0, %1, off"
               :: "v"(lds_off), "v"(gaddr) : "memory");
#endif
}

__device__ __forceinline__ void wait_async_all() {
#if __has_builtin(__builtin_amdgcn_s_wait_asynccnt)
  __builtin_amdgcn_s_wait_asynccnt(0);
#else
  asm volatile("s_wait_asynccnt 0" ::: "memory");
#endif
}

// ---------------- fused warp kernel ----------------
__global__ __launch_bounds__(W_OUT)
void warp_sf_kernel(const float* __restrict__ x,
                    const float* __restrict__ sceneflow,
                    const float* __restrict__ disp,
                    const float* __restrict__ k1,
                    const float* __restrict__ input_size,
                    const int*   __restrict__ ii_ptr,
                    float* __restrict__ out)
{
  __shared__ float sdisp[2][W_IN];        // two source rows of disp
  __shared__ float ssf[3][2][W_IN];       // two source rows x 3 channels of sceneflow

  const int ox = (int)threadIdx.x;        // 0..415 : output column
  const int oy = (int)blockIdx.x;         // 0..127 : output row
  const int cg = (int)blockIdx.y;         // 0..3   : channel group
  const int b  = (int)blockIdx.z;         // batch

  // --- source rows for the 2x bilinear upsample (linspace(0,h-1,oh)) ---
  const float ysf  = (float)oy * ((float)(H_IN - 1) / (float)(H_OUT - 1));
  const int   iy0  = (int)ysf;            // ysf >= 0 -> trunc == floor
  const float fy_r = ysf - (float)iy0;
  const int   iy1  = min(iy0 + 1, H_IN - 1);

  // --- async-stage disp (416 elems) + sceneflow (1248 elems) into LDS ---
  {
    const int t = ox;
    {
      const int r   = t / W_IN;                   // 0 or 1
      const int cc  = t - r * W_IN;
      const int iyr = r ? iy1 : iy0;
      async_copy_b32(disp + ((size_t)b * H_IN + iyr) * W_IN + cc, &sdisp[r][cc]);
    }
#pragma unroll
    for (int k = 0; k < 3; ++k) {
      const int e   = t + k * W_OUT;              // 0..1247
      const int ch  = e / (2 * W_IN);
      const int rem = e - ch * (2 * W_IN);
      const int r   = rem / W_IN;
      const int cc  = rem - r * W_IN;
      const int iyr = r ? iy1 : iy0;
      async_copy_b32(sceneflow + (((size_t)b * 3 + ch) * H_IN + iyr) * W_IN + cc,
                     &ssf[ch][r][cc]);
    }
  }

  // --- scalar camera params (overlapped with async copies in flight) ---
  const float sy = (float)H_OUT / input_size[b * 2 + 0];
  const float sx = (float)W_OUT / input_size[b * 2 + 1];
  const float fx = k1[b * 9 + 0] * sx;
  const float fy = k1[b * 9 + 4] * sy;
  const float cx = k1[b * 9 + 2] * sx;
  const float cy = k1[b * 9 + 5] * sy;
  const int   ii = ii_ptr[0];
  const float coef = 0.625f * (ii == 1) + 1.25f * (ii == 2)
                   + 2.5f   * (ii == 3) + 5.0f  * (ii == 4);

  wait_async_all();
  __syncthreads();

  // --- bilinear 2x upsample of disp / sceneflow from LDS ---
  const float xsf  = (float)ox * ((float)(W_IN - 1) / (float)(W_OUT - 1));
  const int   ix0  = (int)xsf;
  const float fx_r = xsf - (float)ix0;
  const int   ix1  = min(ix0 + 1, W_IN - 1);

  const float w00r = (1.f - fy_r) * (1.f - fx_r);
  const float w01r = (1.f - fy_r) * fx_r;
  const float w10r = fy_r * (1.f - fx_r);
  const float w11r = fy_r * fx_r;

  float dv = sdisp[0][ix0] * w00r + sdisp[0][ix1] * w01r
           + sdisp[1][ix0] * w10r + sdisp[1][ix1] * w11r;
  dv *= (float)W_OUT;

  const float depth = fminf(fmaxf(fx * 0.54f / (dv + 1e-8f), 0.001f), 80.0f);

  const float sf0 = (ssf[0][0][ix0]*w00r + ssf[0][0][ix1]*w01r +
                     ssf[0][1][ix0]*w10r + ssf[0][1][ix1]*w11r) * coef;
  const float sf1 = (ssf[1][0][ix0]*w00r + ssf[1][0][ix1]*w01r +
                     ssf[1][1][ix0]*w10r + ssf[1][1][ix1]*w11r) * coef;
  const float sf2 = (ssf[2][0][ix0]*w00r + ssf[2][0][ix1]*w01r +
                     ssf[2][1][ix0]*w10r + ssf[2][1][ix1]*w11r) * coef;

  // --- back-project (closed-form Kinv), add flow, re-project ---
  const float ptx = ((float)ox - cx) / fx * depth + sf0;
  const float pty = ((float)oy - cy) / fy * depth + sf1;
  const float ptz = depth + sf2;
  const float invw = 1.0f / (ptz + 1e-8f);
  // normalization to [-1,1] and grid_sample's un-normalization cancel exactly:
  const float gx = (fx * ptx + cx * ptz) * invw;
  const float gy = (fy * pty + cy * ptz) * invw;

  // --- bilinear sample weights with zero padding + validity mask ---
  const float x0f = floorf(gx);
  const float y0f = floorf(gy);
  const int sx0 = (int)x0f, sy0 = (int)y0f;
  const int sx1 = sx0 + 1,  sy1 = sy0 + 1;
  const float wxb = gx - x0f, wxa = 1.f - wxb;
  const float wyb = gy - y0f, wya = 1.f - wyb;

  const float vx0 = (sx0 >= 0 && sx0 < W_OUT) ? 1.f : 0.f;
  const float vx1 = (sx1 >= 0 && sx1 < W_OUT) ? 1.f : 0.f;
  const float vy0 = (sy0 >= 0 && sy0 < H_OUT) ? 1.f : 0.f;
  const float vy1 = (sy1 >= 0 && sy1 < H_OUT) ? 1.f : 0.f;

  float w00 = wya * wxa * (vy0 * vx0);
  float w01 = wya * wxb * (vy0 * vx1);
  float w10 = wyb * wxa * (vy1 * vx0);
  float w11 = wyb * wxb * (vy1 * vx1);

  const float msum = w00 + w01 + w10 + w11;        // == grid_sample(ones)
  const float mask = (msum >= 1.0f) ? 1.f : 0.f;   // fold mask into weights
  w00 *= mask; w01 *= mask; w10 *= mask; w11 *= mask;

  const int cxa = min(max(sx0, 0), W_OUT - 1);
  const int cxb = min(max(sx1, 0), W_OUT - 1);
  const int cya = min(max(sy0, 0), H_OUT - 1);
  const int cyb = min(max(sy1, 0), H_OUT - 1);
  const int off00 = cya * W_OUT + cxa;
  const int off01 = cya * W_OUT + cxb;
  const int off10 = cyb * W_OUT + cxa;
  const int off11 = cyb * W_OUT + cxb;

  // --- gather 4 taps per channel; coords amortized over the channel group.
  //     Loads stay default (RT -> x remains L2-resident, 109MB < 192MB L2);
  //     stores are non-temporal so the write-once output doesn't evict x. ---
  const size_t plane = (size_t)HW_OUT;
  const float* xb = x   + ((size_t)b * C_X + (size_t)cg * CPG) * plane;
  float*       ob = out + ((size_t)b * C_X + (size_t)cg * CPG) * plane
                        + (size_t)oy * W_OUT + (size_t)ox;
#pragma unroll 8
  for (int c = 0; c < CPG; ++c) {
    const float* p = xb + (size_t)c * plane;
    const float r = w00 * p[off00] + w01 * p[off01]
                  + w10 * p[off10] + w11 * p[off11];
    __builtin_nontemporal_store(r, ob + (size_t)c * plane);
  }
}

extern "C" void kernel_launch(void* const* d_in, const int* in_sizes, int n_in,
                              void* d_out, int out_size, void* d_ws, size_t ws_size,
                              hipStream_t stream) {
  (void)n_in; (void)out_size; (void)d_ws; (void)ws_size;
  const float* x      = (const float*)d_in[0];
  const float* sf     = (const float*)d_in[1];
  const float* dp     = (const float*)d_in[2];
  const float* k1     = (const float*)d_in[3];
  const float* insz   = (const float*)d_in[4];
  const int*   ii     = (const int*)d_in[5];
  float*       out    = (float*)d_out;

  const int B = in_sizes[3] / 9;            // k1 is (B,3,3)
  dim3 grid(H_OUT, CGRP, B);                // 128 rows x 4 chan-groups x B
  dim3 block(W_OUT, 1, 1);                  // 416 = 13 wave32

  warp_sf_kernel<<<grid, block, 0, stream>>>(x, sf, dp, k1, insz, ii, out);
}